// AdaptiveFourier1d_59296318489247
// MI455X (gfx1250) — compile-verified
//
#include <hip/hip_runtime.h>
#include <math.h>
#include <stdint.h>

// ---------------------------------------------------------------------------
// AFNO-1D on gfx1250.  rfft(4096) -> block-diagonal complex MLP -> irfft.
// All heavy math on V_WMMA_F32_16X16X4_F32; data staged into LDS via the
// CDNA5 Tensor Data Mover (tensor_load_to_lds) and async global->LDS copies
// (global_load_async_to_lds_b128), per CDNA5 ISA.
// 4096-pt FFT = 64x64 Cooley-Tukey: two 64x64x64 GEMM stages vs the 64-pt
// DFT matrix with in-register sincos twiddles.
// ---------------------------------------------------------------------------

typedef float v2f __attribute__((ext_vector_type(2)));
typedef float v8f __attribute__((ext_vector_type(8)));
typedef unsigned int v4u __attribute__((ext_vector_type(4)));
typedef int v4i __attribute__((ext_vector_type(4)));
typedef int v8i __attribute__((ext_vector_type(8)));

#define HIDDEN   768
#define NBLK     8
#define BS       96              // block size (and OB, HSF==1)
#define NN       4096
#define BATCH    16
#define M_MODES  2049            // N/2+1
#define MP       2064            // padded modes: 16*129
#define ROWS     (BATCH*HIDDEN)  // 12288 FFT rows
#define LAM      0.01f
#define TWO_PI   6.28318530717958647692f

// D = A(16x4) * B(4x16) + C(16x16), f32.
// A: lane l holds row (l&15); .x -> K=k0+2*(l>>4), .y -> +1.  B mirrors with
// col (l&15).  C/D: vgpr v, lane l -> row v+8*(l>>4), col l&15.
__device__ __forceinline__ v8f wmma4(v2f a, v2f b, v8f c) {
  return __builtin_amdgcn_wmma_f32_16x16x4_f32(false, a, false, b,
                                               (short)0, c, false, false);
}

__device__ __forceinline__ float softshrink(float v) {
  float t = fmaxf(fabsf(v) - LAM, 0.0f);
  return copysignf(t, v);
}

// Async DMA of 16 bytes/lane from global to LDS (ASYNCcnt-tracked).
__device__ __forceinline__ void async_copy_b128(void* lds_dst, const void* gsrc) {
  uint32_t lds_off = (uint32_t)(uintptr_t)lds_dst;   // LDS aperture: low 32 bits
  asm volatile("global_load_async_to_lds_b128 %0, %1, off"
               :: "v"(lds_off), "v"((uint64_t)(uintptr_t)gsrc)
               : "memory");
}

// TDM: DMA a contiguous run of `elems` f32 from global to LDS (TENSORcnt).
// Descriptor layout per CDNA5 ISA.
// 6-arg builtin form (clang-23 / therock headers): groups 0..3 + extra + cpol.
__device__ __forceinline__ void tdm_load_row_f32(void* lds_dst, const void* gsrc,
                                                 unsigned elems) {
  uint64_t ga = (uint64_t)(uintptr_t)gsrc;
  uint32_t la = (uint32_t)(uintptr_t)lds_dst;
  v4u g0 = {0u, 0u, 0u, 0u};
  g0.x = 1u;                                              // count=1, user D#
  g0.y = la;                                              // lds_addr (bytes)
  g0.z = (uint32_t)ga;                                    // global_addr[31:0]
  g0.w = (uint32_t)((ga >> 32) & 0x01FFFFFFu) | (2u << 30); // [56:32] | type=2
  v8i g1 = {0, 0, 0, 0, 0, 0, 0, 0};
  g1[0] = (int)(2u << 16);                                // data_size = 4B
  g1[1] = (int)((elems & 0xFFFFu) << 16);                 // tensor_dim0 lo16
  g1[2] = (int)((elems >> 16) & 0xFFFFu) | (int)(1u << 16); // td0 hi16 | tensor_dim1=1
  g1[3] = (int)((elems & 0xFFFFu) << 16);                 // tile_dim0
  g1[4] = 1;                                              // tile_dim1 = 1
  g1[5] = (int)elems;                                     // tensor_dim0_stride
  v4i gz4 = {0, 0, 0, 0};
  v8i gz8 = {0, 0, 0, 0, 0, 0, 0, 0};
  __builtin_amdgcn_tensor_load_to_lds(g0, g1, gz4, gz4, gz8, 0);
}

// ===========================================================================
// Kernel 1: forward rfft per row.
//   Stage1: Y[n1][k2] = sum_n2 x[64*n2+n1] * F[n2][k2]      (real x complex)
//   twiddle: Y *= exp(-2pi i n1 k2/4096) / 64               (ortho)
//   Stage2: X[64*k1+k2] = sum_n1 F[k1][n1] * Y[n1][k2]      (complex x complex)
// x row staged to LDS by the Tensor Data Mover (one quarter per wave).
// ===========================================================================
__global__ __launch_bounds__(128) void afno_fft_fwd(const float* __restrict__ x,
                                                    float* __restrict__ xf_re,
                                                    float* __restrict__ xf_im) {
  __shared__ float sX[NN];          // 16 KB, TDM-filled
  __shared__ float sF_re[64 * 64];  // 16 KB
  __shared__ float sF_im[64 * 64];  // 16 KB
  __shared__ float sY_re[64 * 64];  // 16 KB
  __shared__ float sY_im[64 * 64];  // 16 KB

  const int tid  = threadIdx.x;
  const int lane = tid & 31;
  const int wave = tid >> 5;
  const int lrow = lane & 15;
  const int lhi  = lane >> 4;

  const size_t xbase = (size_t)blockIdx.x * NN;

  // TDM: each wave DMAs its 1024-float quarter of the row into LDS.
  tdm_load_row_f32(&sX[wave * 1024], x + xbase + (size_t)wave * 1024, 1024u);

  // Meanwhile build F[j][k] = exp(-2*pi*i*j*k/64) in LDS (symmetric).
  for (int idx = tid; idx < 64 * 64; idx += 128) {
    int j = idx >> 6, k = idx & 63;
    float s, c;
    __sincosf(-TWO_PI * (float)((j * k) & 63) * (1.0f / 64.0f), &s, &c);
    sF_re[idx] = c;
    sF_im[idx] = s;
  }
  __builtin_amdgcn_s_wait_tensorcnt((short)0);
  __syncthreads();

  const int m0 = wave * 16;  // this wave's 16-row output band

  // ---- Stage 1 ----
  for (int ct = 0; ct < 4; ++ct) {
    const int n0 = ct * 16;
    v8f accR = {}, accI = {};
    for (int k0 = 0; k0 < 64; k0 += 4) {
      const int kk = k0 + 2 * lhi;
      v2f a, bR, bI;
      a.x  = sX[64 * kk + (m0 + lrow)];
      a.y  = sX[64 * (kk + 1) + (m0 + lrow)];
      bR.x = sF_re[kk * 64 + (n0 + lrow)];
      bR.y = sF_re[(kk + 1) * 64 + (n0 + lrow)];
      bI.x = sF_im[kk * 64 + (n0 + lrow)];
      bI.y = sF_im[(kk + 1) * 64 + (n0 + lrow)];
      accR = wmma4(a, bR, accR);
      accI = wmma4(a, bI, accI);
    }
    for (int v = 0; v < 8; ++v) {
      int r = m0 + v + 8 * lhi;   // n1
      int c = n0 + lrow;          // k2
      float s, co;
      __sincosf(-TWO_PI * (float)(r * c) * (1.0f / 4096.0f), &s, &co);
      sY_re[r * 64 + c] = (accR[v] * co - accI[v] * s) * 0.015625f;
      sY_im[r * 64 + c] = (accR[v] * s + accI[v] * co) * 0.015625f;
    }
  }
  __syncthreads();

  // ---- Stage 2 ----
  const size_t obase = (size_t)blockIdx.x * MP;
  for (int ct = 0; ct < 4; ++ct) {
    const int n0 = ct * 16;
    v8f pRR = {}, pII = {}, pIM = {};
    for (int k0 = 0; k0 < 64; k0 += 4) {
      const int kk  = k0 + 2 * lhi;
      const int arw = (m0 + lrow) * 64;
      v2f aR, aI, bR, bI;
      aR.x = sF_re[arw + kk];     aR.y = sF_re[arw + kk + 1];
      aI.x = sF_im[arw + kk];     aI.y = sF_im[arw + kk + 1];
      bR.x = sY_re[kk * 64 + (n0 + lrow)];
      bR.y = sY_re[(kk + 1) * 64 + (n0 + lrow)];
      bI.x = sY_im[kk * 64 + (n0 + lrow)];
      bI.y = sY_im[(kk + 1) * 64 + (n0 + lrow)];
      pRR = wmma4(aR, bR, pRR);   // Fre*Yre
      pII = wmma4(aI, bI, pII);   // Fim*Yim
      pIM = wmma4(aR, bI, pIM);   // Fre*Yim
      pIM = wmma4(aI, bR, pIM);   //  + Fim*Yre
    }
    for (int v = 0; v < 8; ++v) {
      int k1 = m0 + v + 8 * lhi;
      int k  = 64 * k1 + (n0 + lrow);
      if (k < M_MODES) {
        xf_re[obase + k] = pRR[v] - pII[v];
        xf_im[obase + k] = pIM[v];
      }
    }
  }
}

// ===========================================================================
// Kernel 2: block-diagonal complex MLP per mode.
// One WG = 4 waves, one block index, 16 column slabs (256 (batch,mode) cols).
// Layer-2 weights staged in LDS once per WG; h bounced per-wave through LDS.
// Complex GEMM = 4 real WMMAs/K-step, separate accumulators (f32 WMMA has no
// A/B negate modifiers).
// ===========================================================================
__global__ __launch_bounds__(128) void afno_mlp(
    const float* __restrict__ w1_re, const float* __restrict__ w1_im,
    const float* __restrict__ w2_re, const float* __restrict__ w2_im,
    const float* __restrict__ b1_re, const float* __restrict__ b1_im,
    const float* __restrict__ b2_re, const float* __restrict__ b2_im,
    const float* __restrict__ xf_re, const float* __restrict__ xf_im,
    float* __restrict__ o_re, float* __restrict__ o_im) {
  __shared__ float sW2re[BS * BS];     // 36 KB
  __shared__ float sW2im[BS * BS];     // 36 KB
  __shared__ float sH_re[4][BS * 16];  // 24 KB
  __shared__ float sH_im[4][BS * 16];  // 24 KB

  const int tid  = threadIdx.x;
  const int lane = tid & 31;
  const int wave = tid >> 5;
  const int lrow = lane & 15;
  const int lhi  = lane >> 4;

  const int nblk = blockIdx.x & 7;
  const int wgrp = blockIdx.x >> 3;              // 0..128
  const size_t wbase = (size_t)nblk * BS * BS;   // w[n][i][o] row-major

  __builtin_prefetch(w1_re + wbase + tid * 8, 0, 1);

  // Stage layer-2 weights once per WG (coalesced b128 loads).
  {
    const float4* g2r = (const float4*)(w2_re + wbase);
    const float4* g2i = (const float4*)(w2_im + wbase);
    float4* s2r = (float4*)sW2re;
    float4* s2i = (float4*)sW2im;
    for (int i = tid; i < BS * BS / 4; i += 128) {
      s2r[i] = g2r[i];
      s2i[i] = g2i[i];
    }
  }
  __syncthreads();

  for (int s = 0; s < 4; ++s) {
    const int slab = wgrp * 16 + wave * 4 + s;   // < BATCH*MP/16 = 2064
    const int col  = slab * 16 + lrow;           // flattened (batch,mode)
    const int bb   = col / MP;
    const int mm   = col % MP;
    const size_t xcol = ((size_t)bb * HIDDEN + nblk * BS) * MP + mm;

    // ---- Layer 1: h = CReLU(sum_i xf[i] * w1[n][i][o] + b1) ----
    for (int rt = 0; rt < 6; ++rt) {
      const int r0 = rt * 16;
      v8f pRR = {}, pII = {}, pIM = {};
      for (int k0 = 0; k0 < BS; k0 += 4) {
        const int kk = k0 + 2 * lhi;
        v2f aR, aI, bR, bI;
        aR.x = w1_re[wbase + (size_t)kk * BS + (r0 + lrow)];
        aR.y = w1_re[wbase + (size_t)(kk + 1) * BS + (r0 + lrow)];
        aI.x = w1_im[wbase + (size_t)kk * BS + (r0 + lrow)];
        aI.y = w1_im[wbase + (size_t)(kk + 1) * BS + (r0 + lrow)];
        bR.x = xf_re[xcol + (size_t)kk * MP];
        bR.y = xf_re[xcol + (size_t)(kk + 1) * MP];
        bI.x = xf_im[xcol + (size_t)kk * MP];
        bI.y = xf_im[xcol + (size_t)(kk + 1) * MP];
        pRR = wmma4(aR, bR, pRR);
        pII = wmma4(aI, bI, pII);
        pIM = wmma4(aR, bI, pIM);
        pIM = wmma4(aI, bR, pIM);
      }
      for (int v = 0; v < 8; ++v) {
        int r = r0 + v + 8 * lhi;
        float hr = pRR[v] - pII[v] + b1_re[nblk * BS + r];
        float hi = pIM[v]          + b1_im[nblk * BS + r];
        sH_re[wave][r * 16 + lrow] = fmaxf(hr, 0.0f);
        sH_im[wave][r * 16 + lrow] = fmaxf(hi, 0.0f);
      }
    }

    // ---- Layer 2: o = softshrink(sum_i h[i] * w2[n][i][o] + b2) ----
    for (int rt = 0; rt < 6; ++rt) {
      const int r0 = rt * 16;
      v8f pRR = {}, pII = {}, pIM = {};
      for (int k0 = 0; k0 < BS; k0 += 4) {
        const int kk = k0 + 2 * lhi;
        v2f aR, aI, bR, bI;
        aR.x = sW2re[kk * BS + (r0 + lrow)];
        aR.y = sW2re[(kk + 1) * BS + (r0 + lrow)];
        aI.x = sW2im[kk * BS + (r0 + lrow)];
        aI.y = sW2im[(kk + 1) * BS + (r0 + lrow)];
        bR.x = sH_re[wave][kk * 16 + lrow];
        bR.y = sH_re[wave][(kk + 1) * 16 + lrow];
        bI.x = sH_im[wave][kk * 16 + lrow];
        bI.y = sH_im[wave][(kk + 1) * 16 + lrow];
        pRR = wmma4(aR, bR, pRR);
        pII = wmma4(aI, bI, pII);
        pIM = wmma4(aR, bI, pIM);
        pIM = wmma4(aI, bR, pIM);
      }
      for (int v = 0; v < 8; ++v) {
        int r = r0 + v + 8 * lhi;
        float orv = pRR[v] - pII[v] + b2_re[nblk * BS + r];
        float oiv = pIM[v]          + b2_im[nblk * BS + r];
        size_t oidx = ((size_t)bb * HIDDEN + nblk * BS + r) * MP + mm;
        o_re[oidx] = softshrink(orv);
        o_im[oidx] = softshrink(oiv);
      }
    }
  }
}

// ===========================================================================
// Kernel 3: irfft + residual.  The 2049 stored modes are async-DMA'd to LDS
// (global_load_async_to_lds_b128, ASYNCcnt), Hermitian-mirrored from LDS,
// then two conj-DFT WMMA stages; final stage needs only the real part.
// ===========================================================================
__global__ __launch_bounds__(128) void afno_fft_inv(
    const float* __restrict__ q_re, const float* __restrict__ q_im,
    const float* __restrict__ x, float* __restrict__ y) {
  __shared__ float sF_re[64 * 64];   // conj DFT: +sin
  __shared__ float sF_im[64 * 64];
  __shared__ float sH_re[64 * 64];
  __shared__ float sH_im[64 * 64];
  __shared__ float sS_re[MP];        // staged spectrum (modes 0..2048, padded)
  __shared__ float sS_im[MP];

  const int tid  = threadIdx.x;
  const int lane = tid & 31;
  const int wave = tid >> 5;
  const int lrow = lane & 15;
  const int lhi  = lane >> 4;

  const size_t sbase = (size_t)blockIdx.x * MP;

  // Async global->LDS staging of the spectrum (b128 = 4 floats per op).
  for (int i = tid; i < MP / 4; i += 128) {
    async_copy_b128(&sS_re[i * 4], q_re + sbase + (size_t)i * 4);
    async_copy_b128(&sS_im[i * 4], q_im + sbase + (size_t)i * 4);
  }

  // Build conj DFT matrix while the DMA is in flight.
  for (int idx = tid; idx < 64 * 64; idx += 128) {
    int j = idx >> 6, k = idx & 63;
    float s, c;
    __sincosf(TWO_PI * (float)((j * k) & 63) * (1.0f / 64.0f), &s, &c);
    sF_re[idx] = c;
    sF_im[idx] = s;
  }
  asm volatile("s_wait_asynccnt 0x0" ::: "memory");
  __syncthreads();

  const int m0 = wave * 16;

  // ---- Stage 1: G[n2][b] = sum_a Fc[n2][a] * S[64a+b]  (Hermitian S) ----
  for (int ct = 0; ct < 4; ++ct) {
    const int n0 = ct * 16;
    v8f pRR = {}, pII = {}, pIM = {};
    for (int k0 = 0; k0 < 64; k0 += 4) {
      const int kk  = k0 + 2 * lhi;
      const int arw = (m0 + lrow) * 64;
      v2f aR, aI, bR, bI;
      aR.x = sF_re[arw + kk];   aR.y = sF_re[arw + kk + 1];
      aI.x = sF_im[arw + kk];   aI.y = sF_im[arw + kk + 1];
      int kA = 64 * kk + (n0 + lrow);
      int kB = kA + 64;
      if (kA < M_MODES) { bR.x = sS_re[kA]; bI.x = sS_im[kA]; }
      else { int km = NN - kA; bR.x = sS_re[km]; bI.x = -sS_im[km]; }
      if (kB < M_MODES) { bR.y = sS_re[kB]; bI.y = sS_im[kB]; }
      else { int km = NN - kB; bR.y = sS_re[km]; bI.y = -sS_im[km]; }
      pRR = wmma4(aR, bR, pRR);
      pII = wmma4(aI, bI, pII);
      pIM = wmma4(aR, bI, pIM);
      pIM = wmma4(aI, bR, pIM);
    }
    for (int v = 0; v < 8; ++v) {
      int r = m0 + v + 8 * lhi;   // n2
      int c = n0 + lrow;          // b
      float s, co;
      __sincosf(TWO_PI * (float)(r * c) * (1.0f / 4096.0f), &s, &co);
      float gr = pRR[v] - pII[v], gi = pIM[v];
      sH_re[r * 64 + c] = (gr * co - gi * s) * 0.015625f;
      sH_im[r * 64 + c] = (gr * s + gi * co) * 0.015625f;
    }
  }
  __syncthreads();

  // ---- Stage 2 (real out): y[64*n1+n2] = Re sum_b Fc[n1][b] * H[n2][b] ----
  const size_t xbase = (size_t)blockIdx.x * NN;
  for (int ct = 0; ct < 4; ++ct) {
    const int n0 = ct * 16;
    v8f p1 = {}, p2 = {};
    for (int k0 = 0; k0 < 64; k0 += 4) {
      const int kk  = k0 + 2 * lhi;
      const int arw = (m0 + lrow) * 64;
      const int brw = (n0 + lrow) * 64;
      v2f aR, aI, bR, bI;
      aR.x = sF_re[arw + kk];   aR.y = sF_re[arw + kk + 1];
      aI.x = sF_im[arw + kk];   aI.y = sF_im[arw + kk + 1];
      bR.x = sH_re[brw + kk];   bR.y = sH_re[brw + kk + 1];
      bI.x = sH_im[brw + kk];   bI.y = sH_im[brw + kk + 1];
      p1 = wmma4(aR, bR, p1);   // Re*Re
      p2 = wmma4(aI, bI, p2);   // Im*Im
    }
    for (int v = 0; v < 8; ++v) {
      int n1 = m0 + v + 8 * lhi;
      int n  = 64 * n1 + (n0 + lrow);
      y[xbase + n] = (p1[v] - p2[v]) + x[xbase + n];
    }
  }
}

// ===========================================================================
extern "C" void kernel_launch(void* const* d_in, const int* in_sizes, int n_in,
                              void* d_out, int out_size, void* d_ws, size_t ws_size,
                              hipStream_t stream) {
  (void)in_sizes; (void)n_in; (void)out_size; (void)ws_size;
  const float* x     = (const float*)d_in[0];
  const float* w1_re = (const float*)d_in[1];
  const float* w1_im = (const float*)d_in[2];
  const float* w2_re = (const float*)d_in[3];
  const float* w2_im = (const float*)d_in[4];
  const float* b1_re = (const float*)d_in[5];
  const float* b1_im = (const float*)d_in[6];
  const float* b2_re = (const float*)d_in[7];
  const float* b2_im = (const float*)d_in[8];
  float* out = (float*)d_out;

  const size_t S = (size_t)ROWS * MP;  // padded spectrum plane
  float* xf_re = (float*)d_ws;
  float* xf_im = xf_re + S;
  float* q_re  = xf_im + S;
  float* q_im  = q_re + S;

  afno_fft_fwd<<<dim3(ROWS), dim3(128), 0, stream>>>(x, xf_re, xf_im);
  afno_mlp<<<dim3(NBLK * (BATCH * MP / 256)), dim3(128), 0, stream>>>(
      w1_re, w1_im, w2_re, w2_im, b1_re, b1_im, b2_re, b2_im,
      xf_re, xf_im, q_re, q_im);
  afno_fft_inv<<<dim3(ROWS), dim3(128), 0, stream>>>(q_re, q_im, x, out);
}